// RankGNN_37684043055385
// MI455X (gfx1250) — compile-verified
//
#include <hip/hip_runtime.h>

typedef __attribute__((ext_vector_type(2))) float v2f;
typedef __attribute__((ext_vector_type(8))) float v8f;

// ---------------------------------------------------------------------------
// WMMA f32 GEMM:  C[N x P] = A[N x KREAL] * B[KREAL x P]
//                 Cagg[i][f] = dinv[i]^2 * C[i][f]   (fused self-loop init)
// One wave computes a 16x16 tile; K swept 4 at a time with
// V_WMMA_F32_16X16X4_F32.  Weights staged zero-padded in LDS in
// pair-interleaved layout (one aligned ds_load_b64 per B fragment).
// A fragments loaded branch-free (clamped index + multiplicative mask).
// Epilogue: wave-uniform full-tile fast path -> unconditional stores.
// ---------------------------------------------------------------------------
template <int KREAL, int KPAD, int P>
__global__ void k_gemm_wmma(const float* __restrict__ A,
                            const float* __restrict__ Bw,
                            const float* __restrict__ dinv,
                            float* __restrict__ C,
                            float* __restrict__ Cagg,
                            int N, int numTiles) {
  constexpr int KP2 = KPAD / 2;
  __shared__ float Bl[KP2][P][2];  // [k-pair][col][k within pair], zero-padded

  {  // cooperative zero-padded weight staging (whole block)
    const int t = threadIdx.y * 32 + threadIdx.x;
    const int nthr = blockDim.y * 32;
    for (int i = t; i < KP2 * P; i += nthr) {
      const int kp = i / P;
      const int cc = i % P;
      const int k0 = 2 * kp;
      Bl[kp][cc][0] = (k0     < KREAL) ? Bw[(size_t)k0 * P + cc]       : 0.0f;
      Bl[kp][cc][1] = (k0 + 1 < KREAL) ? Bw[(size_t)(k0 + 1) * P + cc] : 0.0f;
    }
  }
  __syncthreads();

  const int tile = blockIdx.x * blockDim.y + threadIdx.y;  // wave-uniform
  if (tile >= numTiles) return;
  constexpr int TN = P / 16;
  const int tm = tile / TN;
  const int tn = tile % TN;
  const int lane = threadIdx.x;       // 0..31
  const int half = lane >> 4;         // 0 or 1
  const int l16  = lane & 15;
  const int row  = tm * 16 + l16;     // A row this lane loads
  const bool rv  = row < N;
  const int rr   = rv ? row : 0;      // clamped, always in-bounds
  const int col  = tn * 16 + l16;     // B/C column this lane owns
  const float rm = rv ? 1.0f : 0.0f;

  v8f c = {};
#pragma unroll
  for (int k0 = 0; k0 < KPAD; k0 += 4) {
    const int ka = k0 + 2 * half;
    int  k0a = ka, k1a = ka + 1;
    float m0 = rm, m1 = rm;
    if (KREAL != KPAD) {  // compile-time: only the padded (conv1) GEMM
      if (k0a >= KREAL) { k0a = 0; m0 = 0.0f; }
      if (k1a >= KREAL) { k1a = 0; m1 = 0.0f; }
    }
    v2f a, b;
    a[0] = A[(size_t)rr * KREAL + k0a] * m0;   // contiguous pair when KREAL==KPAD
    a[1] = A[(size_t)rr * KREAL + k1a] * m1;
    b = *(const v2f*)&Bl[(k0 >> 1) + half][col][0];  // one ds_load_b64
    c = __builtin_amdgcn_wmma_f32_16x16x4_f32(
        /*neg_a=*/false, a, /*neg_b=*/false, b,
        /*c_mod=*/(short)0, c, /*reuse_a=*/false, /*reuse_b=*/false);
  }

  if (tm * 16 + 15 < N) {  // wave-uniform fast path: unconditional stores
#pragma unroll
    for (int r = 0; r < 8; ++r) {
      const int orow = tm * 16 + r + 8 * half;
      const float di = dinv[orow];
      const size_t o = (size_t)orow * P + col;
      C[o] = c[r];
      Cagg[o] = di * di * c[r];
    }
  } else {
#pragma unroll
    for (int r = 0; r < 8; ++r) {
      const int orow = tm * 16 + r + 8 * half;
      if (orow < N) {
        const float di = dinv[orow];
        const size_t o = (size_t)orow * P + col;
        C[o] = c[r];
        Cagg[o] = di * di * c[r];
      }
    }
  }
}

// ---------------------------------------------------------------------------
// Graph / elementwise kernels
// ---------------------------------------------------------------------------
__global__ void k_fill1(float* __restrict__ p, long n) {
  long i = blockIdx.x * (long)blockDim.x + threadIdx.x;
  if (i < n) p[i] = 1.0f;
}

__global__ void k_deg_edges(const int* __restrict__ dst, float* __restrict__ deg, long E) {
  long e = blockIdx.x * (long)blockDim.x + threadIdx.x;
  if (e < E) atomicAdd(&deg[dst[e]], 1.0f);
}

__global__ void k_dinv(const float* __restrict__ deg, float* __restrict__ dinv, long n) {
  long i = blockIdx.x * (long)blockDim.x + threadIdx.x;
  if (i >= n) return;
  float d = deg[i];
  dinv[i] = d > 0.0f ? rsqrtf(d) : 0.0f;
}

// agg[dst][f] += dinv[src]*dinv[dst] * h[src][f], float4 per thread
template <int F>
__global__ void k_scatter(const float* __restrict__ h, const int* __restrict__ src,
                          const int* __restrict__ dst, const float* __restrict__ dinv,
                          float* __restrict__ agg, long E) {
  constexpr int F4 = F / 4;
  long idx = blockIdx.x * (long)blockDim.x + threadIdx.x;
  if (idx >= E * F4) return;
  long e = idx / F4;
  int g = (int)(idx % F4);
  int s = src[e];
  int d = dst[e];
  float nrm = dinv[s] * dinv[d];
  const float4 hv = *(const float4*)(h + (size_t)s * F + g * 4);
  float* o = agg + (size_t)d * F + g * 4;
  atomicAdd(o + 0, nrm * hv.x);
  atomicAdd(o + 1, nrm * hv.y);
  atomicAdd(o + 2, nrm * hv.z);
  atomicAdd(o + 3, nrm * hv.w);
}

template <int F>
__global__ void k_bias_tanh(float* __restrict__ buf, const float* __restrict__ bias, int N) {
  long i = blockIdx.x * (long)blockDim.x + threadIdx.x;
  if (i >= (long)N * F) return;
  buf[i] = tanhf(buf[i] + bias[(int)(i & (F - 1))]);
}

// wfc[i] = sum_j Wf1[i][j]*Wf2[j] ; wfc[32] = bf1.Wf2 + bf2
__global__ void k_fusew(const float* __restrict__ Wf1, const float* __restrict__ bf1,
                        const float* __restrict__ Wf2, const float* __restrict__ bf2,
                        float* __restrict__ wfc) {
  int i = threadIdx.x;
  if (i < 32) {
    float s = 0.0f;
#pragma unroll
    for (int j = 0; j < 32; ++j) s += Wf1[i * 32 + j] * Wf2[j];
    wfc[i] = s;
    if (i == 0) {
      float c0 = bf2[0];
      for (int j = 0; j < 32; ++j) c0 += bf1[j] * Wf2[j];
      wfc[32] = c0;
    }
  }
}

// fused: h2 = tanh(agg2 + b2); scalar = h2 . wfc + c0; segment pool atomics
__global__ void k_pool(const float* __restrict__ agg2, const int* __restrict__ batch,
                       const float* __restrict__ wfc, const float* __restrict__ b2,
                       float* __restrict__ sums, float* __restrict__ cnts, int N) {
  int i = blockIdx.x * blockDim.x + threadIdx.x;
  if (i >= N) return;
  const float4* r4 = (const float4*)(agg2 + (size_t)i * 32);
  float s = wfc[32];
#pragma unroll
  for (int q = 0; q < 8; ++q) {
    float4 a = r4[q];
    s += tanhf(a.x + b2[q * 4 + 0]) * wfc[q * 4 + 0] +
         tanhf(a.y + b2[q * 4 + 1]) * wfc[q * 4 + 1] +
         tanhf(a.z + b2[q * 4 + 2]) * wfc[q * 4 + 2] +
         tanhf(a.w + b2[q * 4 + 3]) * wfc[q * 4 + 3];
  }
  int g = batch[i];
  atomicAdd(&sums[g], s);
  atomicAdd(&cnts[g], 1.0f);
}

__global__ void k_util(const float* __restrict__ sums, const float* __restrict__ cnts,
                       float* __restrict__ util, float* __restrict__ out_xutil, int G) {
  int g = blockIdx.x * blockDim.x + threadIdx.x;
  if (g >= G) return;
  float u = sums[g] / fmaxf(cnts[g], 1.0f);
  util[g] = u;
  out_xutil[g] = u;
}

__global__ void k_diff(const float* __restrict__ util, const int* __restrict__ ia,
                       const int* __restrict__ ib, float* __restrict__ out, int NP) {
  int i = blockIdx.x * blockDim.x + threadIdx.x;
  if (i >= NP) return;
  out[i] = util[ib[i]] - util[ia[i]];
}

// ---------------------------------------------------------------------------
extern "C" void kernel_launch(void* const* d_in, const int* in_sizes, int n_in,
                              void* d_out, int out_size, void* d_ws, size_t ws_size,
                              hipStream_t stream) {
  const float* x    = (const float*)d_in[0];
  const int*   ei   = (const int*)d_in[1];
  const int*   batch= (const int*)d_in[2];
  const int*   idxa = (const int*)d_in[3];
  const int*   idxb = (const int*)d_in[4];
  const float* W1   = (const float*)d_in[5];
  const float* b1   = (const float*)d_in[6];
  const float* W2   = (const float*)d_in[7];
  const float* b2   = (const float*)d_in[8];
  const float* Wf1  = (const float*)d_in[9];
  const float* bf1  = (const float*)d_in[10];
  const float* Wf2  = (const float*)d_in[11];
  const float* bf2  = (const float*)d_in[12];
  (void)n_in; (void)ws_size;

  const int  N  = in_sizes[0] / 9;       // 100000
  const long E  = in_sizes[1] / 2;       // 1600000
  const int  NP = in_sizes[3];           // 4096 pairs
  const int  G  = out_size - NP;         // 1024 graphs
  const int* src  = ei;
  const int* dstv = ei + E;

  // workspace carve (256B aligned)
  char* p = (char*)d_ws;
  auto carve = [&](size_t bytes) -> float* {
    char* r = p;
    p += (bytes + 255) & ~(size_t)255;
    return (float*)r;
  };
  float* bufA = carve((size_t)N * 64 * sizeof(float));
  float* bufB = carve((size_t)N * 64 * sizeof(float));
  float* deg  = carve((size_t)N * sizeof(float));
  float* dinv = carve((size_t)N * sizeof(float));
  float* sums = carve((size_t)G * sizeof(float));
  float* cnts = carve((size_t)G * sizeof(float));
  float* util = carve((size_t)G * sizeof(float));
  float* wfc  = carve(64 * sizeof(float));

  auto cdiv = [](long a, long b) { return (int)((a + b - 1) / b); };

  hipMemsetAsync(sums, 0, (size_t)G * sizeof(float), stream);
  hipMemsetAsync(cnts, 0, (size_t)G * sizeof(float), stream);

  // degrees (self-loops included) and rsqrt, shared by both conv layers
  k_fill1<<<cdiv(N, 256), 256, 0, stream>>>(deg, N);
  k_deg_edges<<<cdiv(E, 256), 256, 0, stream>>>(dstv, deg, E);
  k_dinv<<<cdiv(N, 256), 256, 0, stream>>>(deg, dinv, N);

  // ---- conv1: h1 = tanh( Agg( x @ W1 ) + b1 ), F = 64 ----
  {
    const int tiles = ((N + 15) / 16) * (64 / 16);
    k_gemm_wmma<9, 12, 64><<<cdiv(tiles, 8), dim3(32, 8), 0, stream>>>(
        x, W1, dinv, bufA, bufB, N, tiles);  // bufA = xW1, bufB = dinv^2*xW1
  }
  k_scatter<64><<<cdiv(E * 16, 256), 256, 0, stream>>>(bufA, src, dstv, dinv, bufB, E);
  k_bias_tanh<64><<<cdiv((long)N * 64, 256), 256, 0, stream>>>(bufB, b1, N);

  // ---- conv2: h2 = tanh( Agg( h1 @ W2 ) + b2 ), F = 32 ----
  {
    const int tiles = ((N + 15) / 16) * (32 / 16);
    k_gemm_wmma<64, 64, 32><<<cdiv(tiles, 8), dim3(32, 8), 0, stream>>>(
        bufB, W2, dinv, bufA, bufB, N, tiles);  // bufA = h1W2, bufB = dinv^2*h1W2
  }
  k_scatter<32><<<cdiv(E * 8, 256), 256, 0, stream>>>(bufA, src, dstv, dinv, bufB, E);

  // ---- fused: tanh(+b2) o FC (linear o linear collapses) + mean pool + diff ----
  k_fusew<<<1, 32, 0, stream>>>(Wf1, bf1, Wf2, bf2, wfc);
  k_pool<<<cdiv(N, 256), 256, 0, stream>>>(bufB, batch, wfc, b2, sums, cnts, N);
  k_util<<<cdiv(G, 256), 256, 0, stream>>>(sums, cnts, util, (float*)d_out + NP, G);
  k_diff<<<cdiv(NP, 256), 256, 0, stream>>>(util, idxa, idxb, (float*)d_out, NP);
}